// NodeEdgeBlock_29016799052523
// MI455X (gfx1250) — compile-verified
//
#include <hip/hip_runtime.h>

#define BS 8
#define N_ 128
#define DX 256
#define DE 64
#define DY 64

typedef __attribute__((ext_vector_type(2))) float v2f;
typedef __attribute__((ext_vector_type(8))) float v8f;

__device__ __forceinline__ v8f wmma4(v2f a, v2f b, v8f c) {
  // V_WMMA_F32_16X16X4_F32: D = A(16x4) x B(4x16) + C(16x16)
  return __builtin_amdgcn_wmma_f32_16x16x4_f32(false, a, false, b, (short)0, c,
                                               false, false);
}

#define NEG_INF (-__builtin_inff())

// CDNA5 async global->LDS copy (16 bytes per lane), tracked by ASYNCcnt.
__device__ __forceinline__ void async_copy16(void* lds_ptr,
                                             const float* gptr) {
  const unsigned lds_addr = (unsigned)(unsigned long long)lds_ptr;
  asm volatile("global_load_async_to_lds_b128 %0, %1, off"
               :
               : "v"(lds_addr), "v"((unsigned long long)gptr)
               : "memory");
}
__device__ __forceinline__ void wait_async0() {
  asm volatile("s_wait_asynccnt 0x0" ::: "memory");
}

// ---------------------------------------------------------------------------
// K2: ye1/ye2/yx1/yx2 = lin(y, *) : (BS,DY) @ (DY,DX)
// ---------------------------------------------------------------------------
__global__ __launch_bounds__(256) void ymix_kernel(
    const float* __restrict__ y,
    const float* __restrict__ yeaw, const float* __restrict__ yeab,
    const float* __restrict__ yemw, const float* __restrict__ yemb,
    const float* __restrict__ yxaw, const float* __restrict__ yxab,
    const float* __restrict__ yxmw, const float* __restrict__ yxmb,
    float* __restrict__ yea, float* __restrict__ yem,
    float* __restrict__ yxa, float* __restrict__ yxm) {
  const int b = blockIdx.x, c = threadIdx.x;
  const float* yb = y + b * DY;
  float s0 = 0.f, s1 = 0.f, s2 = 0.f, s3 = 0.f;
  for (int k = 0; k < DY; ++k) {
    const float yv = yb[k];
    s0 += yv * yeaw[k * DX + c];
    s1 += yv * yemw[k * DX + c];
    s2 += yv * yxaw[k * DX + c];
    s3 += yv * yxmw[k * DX + c];
  }
  yea[b * DX + c] = s0 + yeab[c];
  yem[b * DX + c] = s1 + yemb[c];
  yxa[b * DX + c] = s2 + yxab[c];
  yxm[b * DX + c] = s3 + yxmb[c];
}

// ---------------------------------------------------------------------------
// K1: Q/K/V = (X @ W + b) * mask via WMMA f32 16x16x4.
// One block = one 16-row M tile (4 waves).  Each wave computes Q,K,V for its
// N-tiles with ONE shared A fragment (3 WMMAs per A load).  All pointers are
// uniform kernel params -> global (not flat) memory ops.
// ---------------------------------------------------------------------------
__global__ __launch_bounds__(128) void qkv_kernel(
    const float* __restrict__ X, const int* __restrict__ nm,
    const float* __restrict__ qw, const float* __restrict__ qb,
    const float* __restrict__ kw, const float* __restrict__ kb,
    const float* __restrict__ vw, const float* __restrict__ vb,
    float* __restrict__ Q, float* __restrict__ Kp, float* __restrict__ Vp) {
  __shared__ float sX[16][DX + 4];
  const int m0 = blockIdx.x * 16;
  const int t = threadIdx.x;
  for (int rep = 0; rep < 8; ++rep) {
    const int chunk = t + rep * 128;  // 1024 16B chunks
    const int row = chunk >> 6;
    const int c4 = (chunk & 63) << 2;
    async_copy16(&sX[row][c4], X + (size_t)(m0 + row) * DX + c4);
  }
  wait_async0();
  __syncthreads();

  const int lane = t & 31, wave = t >> 5;
  const int hlf = lane >> 4, ml = lane & 15;

  for (int nt = wave; nt < 16; nt += 4) {
    const int n0 = nt << 4;
    v8f aq = {}, ak = {}, av = {};
    for (int k0 = 0; k0 < DX; k0 += 4) {
      const int ka = k0 + 2 * hlf, kb2 = ka + 1;
      v2f a, bq, bk, bv;
      a.x = sX[ml][ka];
      a.y = sX[ml][kb2];
      bq.x = qw[(size_t)ka * DX + n0 + ml];
      bq.y = qw[(size_t)kb2 * DX + n0 + ml];
      bk.x = kw[(size_t)ka * DX + n0 + ml];
      bk.y = kw[(size_t)kb2 * DX + n0 + ml];
      bv.x = vw[(size_t)ka * DX + n0 + ml];
      bv.y = vw[(size_t)kb2 * DX + n0 + ml];
      aq = wmma4(a, bq, aq);
      ak = wmma4(a, bk, ak);
      av = wmma4(a, bv, av);
    }
#pragma unroll
    for (int r = 0; r < 8; ++r) {
      const int row = m0 + r + 8 * hlf;
      const int col = n0 + ml;
      const float mk = (float)nm[row];
      Q[(size_t)row * DX + col] = (aq[r] + qb[col]) * mk;
      Kp[(size_t)row * DX + col] = (ak[r] + kb[col]) * mk;
      Vp[(size_t)row * DX + col] = (av[r] + vb[col]) * mk;
    }
  }
}

// ---------------------------------------------------------------------------
// K3: fused edge kernel.  One block per (b, i) row, 256 threads (8 waves).
// Per 16-wide j tile: async-stage E/K/V tiles into LDS, E1/E2 WMMA GEMMs
// (K=64), Y tile in LDS, online softmax + weighted-V accumulation, and fused
// FiLM + eow WMMA projection -> newE.  Y never touches HBM.
// ---------------------------------------------------------------------------
__global__ __launch_bounds__(256) void edge_attn_kernel(
    const float* __restrict__ E, const int* __restrict__ nm,
    const float* __restrict__ Q, const float* __restrict__ Kp,
    const float* __restrict__ Vp,
    const float* __restrict__ emw, const float* __restrict__ emb,
    const float* __restrict__ eaw, const float* __restrict__ eab,
    const float* __restrict__ eow, const float* __restrict__ eob,
    const float* __restrict__ yea, const float* __restrict__ yem,
    float* __restrict__ newE, float* __restrict__ wV) {
  __shared__ float sK[16][DX + 4];
  __shared__ float sV[16][DX + 4];
  __shared__ float sY[16][DX + 4];
  __shared__ float sE[16][DE + 4];
  __shared__ float sQ[DX];
  __shared__ float sYa[DX];  // additive FiLM (from yeaw)
  __shared__ float sYm[DX];  // multiplicative FiLM (from yemw)
  __shared__ float sMj[N_];

  const int b = blockIdx.x >> 7;
  const int i = blockIdx.x & 127;
  const int t = threadIdx.x;
  const int lane = t & 31, wave = t >> 5;
  const int hlf = lane >> 4, ml = lane & 15;
  const float rs = 0.17677669529663688f;  // 1/sqrt(DF=32)

  sYa[t] = yea[b * DX + t];
  sYm[t] = yem[b * DX + t];
  sQ[t] = Q[((size_t)(b * N_ + i)) * DX + t];
  if (t < N_) sMj[t] = (float)nm[b * N_ + t];
  __syncthreads();

  const float mi = sMj[i];
  float rm = NEG_INF, rsum = 0.f, racc = 0.f;  // online-softmax state (n = t)

  for (int jt = 0; jt < 8; ++jt) {
    const int j0 = jt * 16;
    __syncthreads();  // previous tile fully consumed
    {
      const int row = t >> 4, cb = (t & 15) << 4;
      const float* kp = Kp + ((size_t)(b * N_ + j0 + row)) * DX + cb;
      const float* vp = Vp + ((size_t)(b * N_ + j0 + row)) * DX + cb;
#pragma unroll
      for (int q = 0; q < 16; q += 4) {
        async_copy16(&sK[row][cb + q], kp + q);
        async_copy16(&sV[row][cb + q], vp + q);
      }
      const int k4 = (t & 15) << 2;
      async_copy16(&sE[row][k4],
                   E + (((size_t)(b * N_ + i)) * N_ + j0 + row) * DE + k4);
    }
    wait_async0();
    __syncthreads();

    // E1/E2 GEMMs: A = E rows (16 x 64), B = emw/eaw (64 x 256).
    v8f a1[2] = {{}, {}}, a2[2] = {{}, {}};
    for (int k0 = 0; k0 < DE; k0 += 4) {
      const int ka = k0 + 2 * hlf, kb2 = ka + 1;
      v2f a;
      a.x = sE[ml][ka];
      a.y = sE[ml][kb2];
#pragma unroll
      for (int tile = 0; tile < 2; ++tile) {
        const int n0 = wave * 32 + tile * 16;
        v2f bm, ba;
        bm.x = emw[(size_t)ka * DX + n0 + ml];
        bm.y = emw[(size_t)kb2 * DX + n0 + ml];
        ba.x = eaw[(size_t)ka * DX + n0 + ml];
        ba.y = eaw[(size_t)kb2 * DX + n0 + ml];
        a1[tile] = wmma4(a, bm, a1[tile]);
        a2[tile] = wmma4(a, ba, a2[tile]);
      }
    }
    // Y[jj][n] = q*k/sqrt(df) * (E1+1) + E2   (E1,E2 masked by em)
#pragma unroll
    for (int tile = 0; tile < 2; ++tile) {
      const int n = wave * 32 + tile * 16 + ml;
      const float qv = sQ[n] * rs;
#pragma unroll
      for (int r = 0; r < 8; ++r) {
        const int jj = r + 8 * hlf;
        const float em = mi * sMj[j0 + jj];
        const float e1 = (a1[tile][r] + emb[n]) * em;
        const float e2 = (a2[tile][r] + eab[n]) * em;
        sY[jj][n] = qv * sK[jj][n] * (e1 + 1.f) + e2;
      }
    }
    __syncthreads();

    // Online softmax over j (axis=2) + weighted-V accumulation; n = t.
    {
      float yv16[16];
      float tm = rm;
#pragma unroll
      for (int jj = 0; jj < 16; ++jj) {
        const float yv = (sMj[j0 + jj] > 0.f) ? sY[jj][t] : NEG_INF;
        yv16[jj] = yv;
        tm = fmaxf(tm, yv);
      }
      if (tm > rm) {
        const float sc = __expf(rm - tm);  // exp(-inf)=0 on first hit
        rsum *= sc;
        racc *= sc;
        rm = tm;
      }
#pragma unroll
      for (int jj = 0; jj < 16; ++jj) {
        const float p = (yv16[jj] == NEG_INF) ? 0.f : __expf(yv16[jj] - rm);
        rsum += p;
        racc += p * sV[jj][t];
      }
    }

    // newE = (ye1 + (ye2+1)*Y) @ eow + eob, masked.  Waves 0..3, N=64.
    if (wave < 4) {
      const int n0 = wave * 16;
      v8f acc = {};
      for (int k0 = 0; k0 < DX; k0 += 4) {
        const int ka = k0 + 2 * hlf, kb2 = ka + 1;
        v2f a, bv;
        a.x = sYa[ka] + (sYm[ka] + 1.f) * sY[ml][ka];
        a.y = sYa[kb2] + (sYm[kb2] + 1.f) * sY[ml][kb2];
        bv.x = eow[(size_t)ka * DE + n0 + ml];
        bv.y = eow[(size_t)kb2 * DE + n0 + ml];
        acc = wmma4(a, bv, acc);
      }
#pragma unroll
      for (int r = 0; r < 8; ++r) {
        const int jj = r + 8 * hlf;
        const int n = n0 + ml;
        const float em = mi * sMj[j0 + jj];
        newE[(((size_t)(b * N_ + i)) * N_ + j0 + jj) * DE + n] =
            (acc[r] + eob[n]) * em;
      }
    }
  }
  wV[((size_t)(b * N_ + i)) * DX + t] = racc / rsum;
}

// ---------------------------------------------------------------------------
// K4: newX = lin(yx1 + (yx2+1)*weighted_V, xow) * mask via WMMA.
// ---------------------------------------------------------------------------
__global__ __launch_bounds__(128) void newx_kernel(
    const float* __restrict__ wV, const int* __restrict__ nm,
    const float* __restrict__ yxa, const float* __restrict__ yxm,
    const float* __restrict__ xow, const float* __restrict__ xob,
    float* __restrict__ newX) {
  __shared__ float sA[16][DX + 4];
  const int m0 = blockIdx.x * 16;
  const int b = m0 >> 7;
  const int t = threadIdx.x;
  for (int rep = 0; rep < 8; ++rep) {
    const int chunk = t + rep * 128;
    const int row = chunk >> 6;
    const int c4 = (chunk & 63) << 2;
    const float4 wv = *(const float4*)(wV + (size_t)(m0 + row) * DX + c4);
    const float4 aa = *(const float4*)(yxa + (size_t)b * DX + c4);
    const float4 mm = *(const float4*)(yxm + (size_t)b * DX + c4);
    sA[row][c4 + 0] = aa.x + (mm.x + 1.f) * wv.x;
    sA[row][c4 + 1] = aa.y + (mm.y + 1.f) * wv.y;
    sA[row][c4 + 2] = aa.z + (mm.z + 1.f) * wv.z;
    sA[row][c4 + 3] = aa.w + (mm.w + 1.f) * wv.w;
  }
  __syncthreads();

  const int lane = t & 31, wave = t >> 5;
  const int hlf = lane >> 4, ml = lane & 15;
  for (int nt = wave; nt < 16; nt += 4) {
    const int n0 = nt << 4;
    v8f acc = {};
    for (int k0 = 0; k0 < DX; k0 += 4) {
      const int ka = k0 + 2 * hlf, kb2 = ka + 1;
      v2f a, bv;
      a.x = sA[ml][ka];
      a.y = sA[ml][kb2];
      bv.x = xow[(size_t)ka * DX + n0 + ml];
      bv.y = xow[(size_t)kb2 * DX + n0 + ml];
      acc = wmma4(a, bv, acc);
    }
#pragma unroll
    for (int r = 0; r < 8; ++r) {
      const int row = m0 + r + 8 * hlf;
      const int col = n0 + ml;
      newX[(size_t)row * DX + col] = (acc[r] + xob[col]) * (float)nm[row];
    }
  }
}

// ---------------------------------------------------------------------------
// K5a: masked pooling stats over X (axis=1).  Thread = channel.
// ---------------------------------------------------------------------------
__global__ __launch_bounds__(256) void xstats_kernel(
    const float* __restrict__ X, const int* __restrict__ nm,
    float* __restrict__ statsX) {  // [BS][4][DX]
  const int b = blockIdx.x, c = threadIdx.x;
  float s = 0.f, sm = 0.f, sq = 0.f, cnt = 0.f;
  float mi = __builtin_inff(), ma = NEG_INF;
  for (int i = 0; i < N_; ++i) {
    const float m = (float)nm[b * N_ + i];
    const float x = X[((size_t)(b * N_ + i)) * DX + c];
    s += x;
    sm += x * m;
    sq += x * x * m;
    cnt += m;
    const float im = 1.f - m;
    mi = fminf(mi, x + 100000.f * im);
    ma = fmaxf(ma, x - 100000.f * im);
  }
  const float mean = s / cnt;
  const float var = (sq - 2.f * mean * sm + mean * mean * cnt) / cnt;
  float* o = statsX + (size_t)b * 4 * DX;
  o[0 * DX + c] = mean;
  o[1 * DX + c] = mi;
  o[2 * DX + c] = ma;
  o[3 * DX + c] = var;
}

// ---------------------------------------------------------------------------
// K5b: masked pooling stats over E (axes 1,2).  4 segments x 64 channels.
// ---------------------------------------------------------------------------
__global__ __launch_bounds__(256) void estats_kernel(
    const float* __restrict__ E, const int* __restrict__ nm,
    float* __restrict__ statsE) {  // [BS][4][DE]
  __shared__ float red[5][4][DE];
  __shared__ float sM[N_];
  const int b = blockIdx.x;
  const int t = threadIdx.x;
  const int c = t & 63, seg = t >> 6;
  if (t < N_) sM[t] = (float)nm[b * N_ + t];
  __syncthreads();

  float s = 0.f, sm = 0.f, sq = 0.f;
  float mi = __builtin_inff(), ma = NEG_INF;
  for (int p = seg * 4096; p < (seg + 1) * 4096; ++p) {
    const float em = sM[p >> 7] * sM[p & 127];
    const float v = E[((size_t)b * 16384 + p) * DE + c];
    s += v;
    sm += v * em;
    sq += v * v * em;
    const float im = 1.f - em;
    mi = fminf(mi, v + 100000.f * im);
    ma = fmaxf(ma, v - 100000.f * im);
  }
  red[0][seg][c] = s;
  red[1][seg][c] = sm;
  red[2][seg][c] = sq;
  red[3][seg][c] = mi;
  red[4][seg][c] = ma;
  __syncthreads();
  if (t < DE) {
    float S = 0.f, SM = 0.f, SQ = 0.f;
    float MI = __builtin_inff(), MA = NEG_INF;
    for (int g = 0; g < 4; ++g) {
      S += red[0][g][t];
      SM += red[1][g][t];
      SQ += red[2][g][t];
      MI = fminf(MI, red[3][g][t]);
      MA = fmaxf(MA, red[4][g][t]);
    }
    float cx = 0.f;
    for (int i = 0; i < N_; ++i) cx += sM[i];
    const float cnt = cx * cx;
    const float mean = S / cnt;
    const float var = (SQ - 2.f * mean * SM + mean * mean * cnt) / cnt;
    float* o = statsE + (size_t)b * 4 * DE;
    o[0 * DE + t] = mean;
    o[1 * DE + t] = MI;
    o[2 * DE + t] = MA;
    o[3 * DE + t] = var;
  }
}

// ---------------------------------------------------------------------------
// K6: y head.
// ---------------------------------------------------------------------------
__global__ __launch_bounds__(64) void yhead_kernel(
    const float* __restrict__ y, const float* __restrict__ statsX,
    const float* __restrict__ statsE, const float* __restrict__ xyw,
    const float* __restrict__ xyb, const float* __restrict__ eyw,
    const float* __restrict__ eyb, const float* __restrict__ yyw,
    const float* __restrict__ yyb, const float* __restrict__ yo1w,
    const float* __restrict__ yo1b, const float* __restrict__ yo2w,
    const float* __restrict__ yo2b, float* __restrict__ new_y) {
  __shared__ float s1[DY], s2[DY];
  const int b = blockIdx.x, c = threadIdx.x;
  float acc = xyb[c] + eyb[c] + yyb[c];
  const float* sx = statsX + (size_t)b * 4 * DX;
  for (int k = 0; k < 4 * DX; ++k) acc += sx[k] * xyw[(size_t)k * DY + c];
  const float* se = statsE + (size_t)b * 4 * DE;
  for (int k = 0; k < 4 * DE; ++k) acc += se[k] * eyw[(size_t)k * DY + c];
  const float* yb = y + b * DY;
  for (int k = 0; k < DY; ++k) acc += yb[k] * yyw[k * DY + c];
  s1[c] = acc;
  __syncthreads();
  float h = yo1b[c];
  for (int k = 0; k < DY; ++k) h += s1[k] * yo1w[k * DY + c];
  s2[c] = fmaxf(h, 0.f);
  __syncthreads();
  float o = yo2b[c];
  for (int k = 0; k < DY; ++k) o += s2[k] * yo2w[k * DY + c];
  new_y[b * DY + c] = o;
}

// ---------------------------------------------------------------------------
extern "C" void kernel_launch(void* const* d_in, const int* in_sizes, int n_in,
                              void* d_out, int out_size, void* d_ws,
                              size_t ws_size, hipStream_t stream) {
  (void)in_sizes; (void)n_in; (void)out_size; (void)ws_size;
  const float* X = (const float*)d_in[0];
  const float* E = (const float*)d_in[1];
  const float* y = (const float*)d_in[2];
  const int* nm = (const int*)d_in[3];
  const float* qw = (const float*)d_in[4], * qb = (const float*)d_in[5];
  const float* kw = (const float*)d_in[6], * kb = (const float*)d_in[7];
  const float* vw = (const float*)d_in[8], * vb = (const float*)d_in[9];
  const float* eaw = (const float*)d_in[10], * eab = (const float*)d_in[11];
  const float* emw = (const float*)d_in[12], * emb = (const float*)d_in[13];
  const float* yemw = (const float*)d_in[14], * yemb = (const float*)d_in[15];
  const float* yeaw = (const float*)d_in[16], * yeab = (const float*)d_in[17];
  const float* yxmw = (const float*)d_in[18], * yxmb = (const float*)d_in[19];
  const float* yxaw = (const float*)d_in[20], * yxab = (const float*)d_in[21];
  const float* yyw = (const float*)d_in[22], * yyb = (const float*)d_in[23];
  const float* xyw = (const float*)d_in[24], * xyb = (const float*)d_in[25];
  const float* eyw = (const float*)d_in[26], * eyb = (const float*)d_in[27];
  const float* xow = (const float*)d_in[28], * xob = (const float*)d_in[29];
  const float* eow = (const float*)d_in[30], * eob = (const float*)d_in[31];
  const float* yo1w = (const float*)d_in[32], * yo1b = (const float*)d_in[33];
  const float* yo2w = (const float*)d_in[34], * yo2b = (const float*)d_in[35];

  float* out = (float*)d_out;
  float* newX = out;                        // 8*128*256
  float* newE = out + 262144;               // 8*128*128*64
  float* new_y = out + 262144 + 8388608;    // 8*64

  float* ws = (float*)d_ws;
  float* Q = ws;                    // 262144
  float* Kb = ws + 262144;          // 262144
  float* Vb = ws + 524288;          // 262144
  float* wV = ws + 786432;          // 262144
  float* yea = ws + 1048576;        // 2048
  float* yem = yea + 2048;
  float* yxa = yem + 2048;
  float* yxm = yxa + 2048;
  float* statsX = yxm + 2048;       // 8192
  float* statsE = statsX + 8192;    // 2048

  ymix_kernel<<<BS, 256, 0, stream>>>(y, yeaw, yeab, yemw, yemb, yxaw, yxab,
                                      yxmw, yxmb, yea, yem, yxa, yxm);
  qkv_kernel<<<64, 128, 0, stream>>>(X, nm, qw, qb, kw, kb, vw, vb, Q, Kb, Vb);
  edge_attn_kernel<<<BS * N_, 256, 0, stream>>>(E, nm, Q, Kb, Vb, emw, emb,
                                                eaw, eab, eow, eob, yea, yem,
                                                newE, wV);
  newx_kernel<<<64, 128, 0, stream>>>(wV, nm, yxa, yxm, xow, xob, newX);
  xstats_kernel<<<BS, 256, 0, stream>>>(X, nm, statsX);
  estats_kernel<<<BS, 256, 0, stream>>>(E, nm, statsE);
  yhead_kernel<<<BS, 64, 0, stream>>>(y, statsX, statsE, xyw, xyb, eyw, eyb,
                                      yyw, yyb, yo1w, yo1b, yo2w, yo2b, new_y);
}